// kabsch_torch_53274774339929
// MI455X (gfx1250) — compile-verified
//
#include <hip/hip_runtime.h>
#include <math.h>

// CDNA5 / gfx1250. Kabsch: one bandwidth-bound streaming reduction pass
// (96 MB @ 23.3 TB/s ~ 4.1us floor), then a tiny 3x3 SVD tail that runs the
// matrix products through V_WMMA_F32_16X16X4_F32 on a single full wave.

typedef float v2f __attribute__((ext_vector_type(2)));
typedef float v8f __attribute__((ext_vector_type(8)));

#define THREADS 256
#define MAXBLOCKS 1024

// ---------------- Pass 1: streaming 15-way reduction ----------------
// partials per block: [0..8] = sum P_i Q_j (outer product), [9..11]=sum P, [12..14]=sum Q
__global__ __launch_bounds__(THREADS) void kabsch_pass1(
    const float* __restrict__ P, const float* __restrict__ Q,
    int n, float* __restrict__ ws)
{
    float s[15];
#pragma unroll
    for (int k = 0; k < 15; ++k) s[k] = 0.0f;

    const int stride = gridDim.x * blockDim.x;
    for (int idx = blockIdx.x * blockDim.x + threadIdx.x; idx < n; idx += stride) {
        const float* pp = P + 3 * (size_t)idx;
        const float* qq = Q + 3 * (size_t)idx;
        // speculative streaming prefetch a few grid-strides ahead (global_prefetch_b8)
        __builtin_prefetch(pp + 3 * (size_t)stride * 4, 0, 0);
        __builtin_prefetch(qq + 3 * (size_t)stride * 4, 0, 0);
        float px = pp[0], py = pp[1], pz = pp[2];
        float qx = qq[0], qy = qq[1], qz = qq[2];
        s[0] = fmaf(px, qx, s[0]); s[1] = fmaf(px, qy, s[1]); s[2] = fmaf(px, qz, s[2]);
        s[3] = fmaf(py, qx, s[3]); s[4] = fmaf(py, qy, s[4]); s[5] = fmaf(py, qz, s[5]);
        s[6] = fmaf(pz, qx, s[6]); s[7] = fmaf(pz, qy, s[7]); s[8] = fmaf(pz, qz, s[8]);
        s[9]  += px; s[10] += py; s[11] += pz;
        s[12] += qx; s[13] += qy; s[14] += qz;
    }

    // deterministic fixed-tree wave32 reduction
#pragma unroll
    for (int k = 0; k < 15; ++k) {
#pragma unroll
        for (int off = 16; off > 0; off >>= 1)
            s[k] += __shfl_down(s[k], off, 32);
    }

    __shared__ float part[THREADS / 32][15];
    const int lane = threadIdx.x & 31;
    const int wv   = threadIdx.x >> 5;
    if (lane == 0) {
#pragma unroll
        for (int k = 0; k < 15; ++k) part[wv][k] = s[k];
    }
    __syncthreads();
    if (threadIdx.x < 15) {
        float tot = 0.0f;
#pragma unroll
        for (int w = 0; w < THREADS / 32; ++w) tot += part[w][threadIdx.x];  // fixed order
        ws[(size_t)blockIdx.x * 15 + threadIdx.x] = tot;
    }
}

// ---------------- 3x3 matmul on the matrix pipe ----------------
// Z = X * Y via V_WMMA_F32_16X16X4_F32 (A:16x4, B:4x16, zero padded).
// Must be called by one full wave32 with EXEC all ones at the WMMA.
__device__ inline void wmma3x3(const float X[9], const float Y[9],
                               float Z[9], float* buf)
{
    const int lane = threadIdx.x & 31;
    const int r  = lane & 15;
    const int hi = lane >> 4;
    v2f a; a.x = 0.0f; a.y = 0.0f;
    v2f b; b.x = 0.0f; b.y = 0.0f;
    if (r < 3) {
        // A-matrix 16x4 f32: lanes 0-15 hold K=0,1 in v0,v1; lanes 16-31 hold K=2,3
        if (hi == 0) { a.x = X[r * 3 + 0]; a.y = X[r * 3 + 1]; }
        else         { a.x = X[r * 3 + 2]; }
        // B-matrix 4x16 f32: lanes 0-15 rows K=0,1; lanes 16-31 rows K=2,3
        if (hi == 0) { b.x = Y[0 + r];     b.y = Y[3 + r]; }
        else         { b.x = Y[6 + r]; }
    }
    v8f c = {};
    c = __builtin_amdgcn_wmma_f32_16x16x4_f32(
            /*neg_a=*/false, a, /*neg_b=*/false, b,
            /*c_mod=*/(short)0, c, /*reuse_a=*/false, /*reuse_b=*/false);
    __syncthreads();
    // D layout: VGPR m holds D[m][lane] for lanes 0-15
    if (lane < 3) {
        buf[0 * 3 + lane] = c[0];
        buf[1 * 3 + lane] = c[1];
        buf[2 * 3 + lane] = c[2];
    }
    __syncthreads();
#pragma unroll
    for (int i = 0; i < 9; ++i) Z[i] = buf[i];
    __syncthreads();
}

// Cyclic Jacobi eigensolver for symmetric 3x3 (fixed sweep count => deterministic)
__device__ inline void jacobi3(const float Sin[9], float V[9], float lam[3])
{
    float A[9];
#pragma unroll
    for (int i = 0; i < 9; ++i) { A[i] = Sin[i]; V[i] = 0.0f; }
    V[0] = V[4] = V[8] = 1.0f;
    const int PQ[3][2] = { {0, 1}, {0, 2}, {1, 2} };
    for (int sweep = 0; sweep < 16; ++sweep) {
        for (int r = 0; r < 3; ++r) {
            const int p = PQ[r][0], q = PQ[r][1];
            float apq = A[p * 3 + q];
            if (fabsf(apq) > 1e-24f) {  // uniform across wave (replicated data)
                float app = A[p * 3 + p], aqq = A[q * 3 + q];
                float tau = (aqq - app) / (2.0f * apq);
                float tt  = copysignf(1.0f, tau) / (fabsf(tau) + sqrtf(1.0f + tau * tau));
                float cc  = 1.0f / sqrtf(1.0f + tt * tt);
                float ss  = tt * cc;
                for (int k = 0; k < 3; ++k) {  // A <- A * G
                    float akp = A[k * 3 + p], akq = A[k * 3 + q];
                    A[k * 3 + p] = cc * akp - ss * akq;
                    A[k * 3 + q] = ss * akp + cc * akq;
                }
                for (int k = 0; k < 3; ++k) {  // A <- G^T * A
                    float apk = A[p * 3 + k], aqk = A[q * 3 + k];
                    A[p * 3 + k] = cc * apk - ss * aqk;
                    A[q * 3 + k] = ss * apk + cc * aqk;
                }
                for (int k = 0; k < 3; ++k) {  // V <- V * G
                    float vkp = V[k * 3 + p], vkq = V[k * 3 + q];
                    V[k * 3 + p] = cc * vkp - ss * vkq;
                    V[k * 3 + q] = ss * vkp + cc * vkq;
                }
            }
        }
    }
    lam[0] = A[0]; lam[1] = A[4]; lam[2] = A[8];
}

__device__ inline float det3(const float M[9])
{
    return M[0] * (M[4] * M[8] - M[5] * M[7])
         - M[1] * (M[3] * M[8] - M[5] * M[6])
         + M[2] * (M[3] * M[7] - M[4] * M[6]);
}

// ---------------- Pass 2: final reduce + 3x3 SVD tail (1 wave) ----------------
__global__ __launch_bounds__(32) void kabsch_pass2(
    const float* __restrict__ ws, int nblocks, int n, float* __restrict__ out)
{
    __shared__ float red[15];
    __shared__ float buf[9];
    const int lane = threadIdx.x;

    if (lane < 15) {
        float tot = 0.0f;
        for (int b = 0; b < nblocks; ++b) tot += ws[(size_t)b * 15 + lane];  // fixed order
        red[lane] = tot;
    }
    __syncthreads();

    float Spq[9], sumP[3], sumQ[3];
#pragma unroll
    for (int i = 0; i < 9; ++i) Spq[i] = red[i];
#pragma unroll
    for (int i = 0; i < 3; ++i) { sumP[i] = red[9 + i]; sumQ[i] = red[12 + i]; }
    const float invN = 1.0f / (float)n;

    float H[9];
#pragma unroll
    for (int i = 0; i < 3; ++i)
#pragma unroll
        for (int j = 0; j < 3; ++j)
            H[i * 3 + j] = Spq[i * 3 + j] - sumP[i] * sumQ[j] * invN;
    float tvec[3];
#pragma unroll
    for (int i = 0; i < 3; ++i) tvec[i] = (sumQ[i] - sumP[i]) * invN;

    // S = H^T H  (matrix pipe)
    float HT[9];
#pragma unroll
    for (int i = 0; i < 3; ++i)
#pragma unroll
        for (int j = 0; j < 3; ++j) HT[i * 3 + j] = H[j * 3 + i];
    float S[9];
    wmma3x3(HT, H, S, buf);

    // Eigendecomposition of S -> right singular vectors V, lam = sigma^2
    float V[9], lam[3];
    jacobi3(S, V, lam);
    // sort descending (SVD convention), swapping V columns
    for (int a = 0; a < 2; ++a)
        for (int b = a + 1; b < 3; ++b)
            if (lam[b] > lam[a]) {
                float tl = lam[a]; lam[a] = lam[b]; lam[b] = tl;
                for (int k = 0; k < 3; ++k) {
                    float tv = V[k * 3 + a]; V[k * 3 + a] = V[k * 3 + b]; V[k * 3 + b] = tv;
                }
            }

    // W = H V = U * Sigma  (matrix pipe); U columns = W columns / sigma
    float W[9];
    wmma3x3(H, V, W, buf);
    float Umat[9];
#pragma unroll
    for (int j = 0; j < 3; ++j) {
        float sg = sqrtf(fmaxf(lam[j], 0.0f));
        float iv = (sg > 1e-20f) ? (1.0f / sg) : 0.0f;
#pragma unroll
        for (int i = 0; i < 3; ++i) Umat[i * 3 + j] = W[i * 3 + j] * iv;
    }

    // M0 = Vt^T U^T = V U^T  (matrix pipe)
    float UT[9];
#pragma unroll
    for (int i = 0; i < 3; ++i)
#pragma unroll
        for (int j = 0; j < 3; ++j) UT[i * 3 + j] = Umat[j * 3 + i];
    float M0[9];
    wmma3x3(V, UT, M0, buf);

    // Reference: sign = det(Vt.T @ U.T) < 0 ? -1 : 1; Vt[:, -1] *= sign; R = Vt.T @ U.T
    // Flipping Vt's last column == scaling row 2 of M0 by sign.
    float sgn = (det3(M0) < 0.0f) ? -1.0f : 1.0f;
#pragma unroll
    for (int j = 0; j < 3; ++j) M0[6 + j] *= sgn;

    if (lane == 0) {
#pragma unroll
        for (int i = 0; i < 9; ++i) out[i] = M0[i];  // R row-major
#pragma unroll
        for (int i = 0; i < 3; ++i) out[9 + i] = tvec[i];  // t
    }
}

extern "C" void kernel_launch(void* const* d_in, const int* in_sizes, int n_in,
                              void* d_out, int out_size, void* d_ws, size_t ws_size,
                              hipStream_t stream)
{
    const float* P = (const float*)d_in[0];
    const float* Q = (const float*)d_in[1];
    float* out = (float*)d_out;
    float* ws  = (float*)d_ws;
    const int n = in_sizes[0] / 3;

    int maxb = (int)(ws_size / (15 * sizeof(float)));
    int nblocks = maxb < MAXBLOCKS ? maxb : MAXBLOCKS;
    if (nblocks < 1) nblocks = 1;

    kabsch_pass1<<<dim3(nblocks), dim3(THREADS), 0, stream>>>(P, Q, n, ws);
    kabsch_pass2<<<dim3(1), dim3(32), 0, stream>>>(ws, nblocks, n, out);
}